// QuasiNewtonMethod_7232724927245
// MI455X (gfx1250) — compile-verified
//
#include <hip/hip_runtime.h>

// ---------------- CDNA5 / gfx1250 specifics ----------------
typedef __attribute__((ext_vector_type(2))) float v2f;   // WMMA f32 A/B chunk (16x4 / 4x16)
typedef __attribute__((ext_vector_type(8))) float v8f;   // WMMA f32 C/D (16x16)

// Wave-local LDS producer->consumer ordering using CDNA5 split wait counters.
// (All LDS traffic here is private to one wave; no block barrier needed.)
#define LDS_SYNC_WAVE()                                   \
  do {                                                    \
    __builtin_amdgcn_wave_barrier();                      \
    asm volatile("s_wait_dscnt 0" ::: "memory");          \
    __builtin_amdgcn_wave_barrier();                      \
  } while (0)

static constexpr int   WAVES   = 8;       // waves (=batches) per block
static constexpr float RO      = 0.5f;    // backtracking shrink
static constexpr float C_ARM   = 1e-4f;   // Armijo constant
static constexpr int   WSTRIDE = 400;     // floats of LDS per wave (256 H-stage + 9 vectors)

// D(16x16) = A(16x16) * broadcast(v) via 4 chained V_WMMA_F32_16X16X4_F32.
// A is held as 4 chunks in the ISA A-layout: lane (m,half) holds
// A[m][4c + 2*half + j], j=0..1.  B broadcasts v into every column, so the
// B chunk is b[j] = v[4c + 2*half + j] (column-independent).
static __device__ __forceinline__ v8f wmma_mv(const v2f A[4], const float* v, int colbase) {
  v8f acc = {};
#pragma unroll
  for (int c = 0; c < 4; ++c) {
    v2f b;
    b[0] = v[4 * c + colbase + 0];
    b[1] = v[4 * c + colbase + 1];
    // 8 args: (neg_a, A, neg_b, B, c_mod, C, reuse_a, reuse_b)
    acc = __builtin_amdgcn_wmma_f32_16x16x4_f32(false, A[c], false, b, (short)0, acc,
                                                false, false);
  }
  return acc;
}

// With broadcast-B every column of D equals A*v.  Lane (m==0) of each half
// commits rows [8*half .. 8*half+7] (C/D layout: VGPR r -> row r + 8*half).
static __device__ __forceinline__ void commit_vec(float* dst, v8f d, int half, int m,
                                                  float scale) {
  if (m == 0) {
    float4 lo = make_float4(scale * d[0], scale * d[1], scale * d[2], scale * d[3]);
    float4 hi = make_float4(scale * d[4], scale * d[5], scale * d[6], scale * d[7]);
    *(float4*)&dst[8 * half + 0] = lo;
    *(float4*)&dst[8 * half + 4] = hi;
  }
}

static __device__ __forceinline__ float wave_sum(float v) {
#pragma unroll
  for (int off = 16; off > 0; off >>= 1) v += __shfl_xor(v, off, 32);
  return v;
}

__global__ __launch_bounds__(WAVES * 32)
void bfgs16_kernel(const float* __restrict__ yg, const float* __restrict__ hg,
                   const float* __restrict__ x0g, const int* __restrict__ itg,
                   float* __restrict__ outg, int B) {
  __shared__ __align__(16) float lds[WAVES * WSTRIDE];

  const int lane  = threadIdx.x & 31;
  const int wid   = threadIdx.x >> 5;
  const int batch = blockIdx.x * WAVES + wid;
  if (batch >= B) return;                 // wave-uniform predicate (EXEC all-1s for WMMA)

  const int m       = lane & 15;
  const int half    = lane >> 4;
  const int colbase = 2 * half;

  float* wl  = lds + wid * WSTRIDE;       // per-wave slice
  float* Yv  = wl + 256;
  float* Xv  = wl + 272;
  float* Gv  = wl + 288;
  float* R0  = wl + 304;
  float* HP  = wl + 320;
  float* Pv  = wl + 336;
  float* YKv = wl + 352;
  float* Sv  = wl + 368;
  float* Uv  = wl + 384;

  // ---- stage H through LDS: two coalesced b128 loads per lane ----
  const float* hb = hg + (size_t)batch * 256;
  {
    const float4* hb4 = (const float4*)hb;
    float4 q0 = hb4[lane * 2 + 0];
    float4 q1 = hb4[lane * 2 + 1];
    ((float4*)wl)[lane * 2 + 0] = q0;
    ((float4*)wl)[lane * 2 + 1] = q1;
  }
  __builtin_prefetch(hb + 256, 0, 1);     // next batch's H -> L2 (global_prefetch_b8)
  if (half == 0) {
    Yv[m] = yg[(size_t)batch * 16 + m];
    Xv[m] = x0g[(size_t)batch * 16 + m];
  }
  LDS_SYNC_WAVE();

  // H, H^T and Hk=I in WMMA A-operand layout (row m, cols 4c+2*half+{0,1}).
  v2f HA[4], HTA[4], Hk[4];
#pragma unroll
  for (int c = 0; c < 4; ++c) {
    const int c0 = 4 * c + colbase;
    HA[c][0]  = wl[m * 16 + c0];
    HA[c][1]  = wl[m * 16 + c0 + 1];
    HTA[c][0] = wl[c0 * 16 + m];
    HTA[c][1] = wl[(c0 + 1) * 16 + m];
    Hk[c][0]  = (m == c0)     ? 1.0f : 0.0f;
    Hk[c][1]  = (m == c0 + 1) ? 1.0f : 0.0f;
  }

  // r0 = y - H x ;  g = H^T(Hx - y) = -H^T r0
  {
    v8f t = wmma_mv(HA, Xv, colbase);
    if (m == 0) {
#pragma unroll
      for (int r = 0; r < 8; ++r) R0[8 * half + r] = Yv[8 * half + r] - t[r];
    }
  }
  LDS_SYNC_WAVE();
  {
    v8f gt = wmma_mv(HTA, R0, colbase);
    commit_vec(Gv, gt, half, m, -1.0f);
  }
  LDS_SYNC_WAVE();

  float alpha = 1.0f;                     // persists across outer iterations (as in ref)
  const int iters = itg[0];

  for (int it = 0; it < iters; ++it) {
    // p = -(Hk g)
    v8f pv = wmma_mv(Hk, Gv, colbase);
    commit_vec(Pv, pv, half, m, -1.0f);
    LDS_SYNC_WAVE();

    // hp = H p  (single extra matvec makes every line-search trial vector-only)
    v8f hpv = wmma_mv(HA, Pv, colbase);
    commit_vec(HP, hpv, half, m, 1.0f);
    LDS_SYNC_WAVE();

    // fx = 0.5*||r0||, gtp = g.p   (wave-uniform scalars)
    float pr = R0[m] * R0[m];
    float pg = Gv[m] * Pv[m];
    if (half) { pr = 0.0f; pg = 0.0f; }
    const float fx  = 0.5f * sqrtf(wave_sum(pr));
    const float gtp = wave_sum(pg);

    // Armijo backtracking: f(x+a p) = 0.5*||r0 - a*hp||  — uniform loop, no divergence
    float a = alpha;
    const float r0m = R0[m], hpm = HP[m];
    for (int ls = 0; ls < 200; ++ls) {
      const float d  = r0m - a * hpm;
      const float sq = half ? 0.0f : d * d;
      const float lhs = 0.5f * sqrtf(wave_sum(sq));
      if (!(lhs > fx + C_ARM * a * gtp)) break;
      a *= RO;
    }
    alpha = a;

    // s = a*p ; x += s ; r0 -= a*hp
    if (half == 0) {
      const float sm = a * Pv[m];
      Sv[m] = sm;
      Xv[m] += sm;
      R0[m] -= a * HP[m];
    }
    LDS_SYNC_WAVE();

    // yk = g(x_new) - g = a * H^T (H p) ;  g += yk
    v8f wv = wmma_mv(HTA, HP, colbase);
    if (m == 0) {
#pragma unroll
      for (int r = 0; r < 8; ++r) {
        const float ykv = a * wv[r];
        YKv[8 * half + r] = ykv;
        Gv[8 * half + r] += ykv;
      }
    }
    LDS_SYNC_WAVE();

    // u = Hk yk
    v8f uv = wmma_mv(Hk, YKv, colbase);
    commit_vec(Uv, uv, half, m, 1.0f);
    LDS_SYNC_WAVE();

    // BFGS scalars and rank-2 update of Hk, done directly in A-layout:
    // Hk[m][col] += k1*s[m]*s[col] - (u[m]*s[col] + s[m]*u[col])*k2
    float d0 = Sv[m] * YKv[m];
    float d1 = YKv[m] * Uv[m];
    if (half) { d0 = 0.0f; d1 = 0.0f; }
    const float auxSc = wave_sum(d0);
    const float ytu   = wave_sum(d1);
    const float k1 = (auxSc + ytu) / (auxSc * auxSc);
    const float k2 = 1.0f / auxSc;

    const float sm = Sv[m], um = Uv[m];
#pragma unroll
    for (int c = 0; c < 4; ++c) {
      const int cc = 4 * c + colbase;
      const float2 sc = *(const float2*)&Sv[cc];
      const float2 uc = *(const float2*)&Uv[cc];
      Hk[c][0] += k1 * sm * sc.x - (um * sc.x + sm * uc.x) * k2;
      Hk[c][1] += k1 * sm * sc.y - (um * sc.y + sm * uc.y) * k2;
    }
    // Same-wave DS ordering guarantees these reads retire before the next
    // iteration's DS writes to Sv/Uv/YKv.
  }

  if (half == 0) outg[(size_t)batch * 16 + m] = Xv[m];
}

extern "C" void kernel_launch(void* const* d_in, const int* in_sizes, int n_in,
                              void* d_out, int out_size, void* d_ws, size_t ws_size,
                              hipStream_t stream) {
  (void)n_in; (void)out_size; (void)d_ws; (void)ws_size;
  const float* y  = (const float*)d_in[0];
  const float* h  = (const float*)d_in[1];
  const float* x0 = (const float*)d_in[2];
  const int*   it = (const int*)d_in[3];
  float* out = (float*)d_out;

  const int B = in_sizes[0] / 16;               // y is [B,16]
  dim3 block(WAVES * 32);
  dim3 grid((B + WAVES - 1) / WAVES);
  bfgs16_kernel<<<grid, block, 0, stream>>>(y, h, x0, it, out, B);
}